// Encoder_80161269612638
// MI455X (gfx1250) — compile-verified
//
#include <hip/hip_runtime.h>
#include <hip/hip_bf16.h>
#include <math.h>

typedef float v2f __attribute__((ext_vector_type(2)));
typedef float v8f __attribute__((ext_vector_type(8)));

#define NB   8
#define NN   768
#define NC   16        // 16 SPHC channels (l = 0..3)
#define RCUT 5.0f
#define EPSV 1e-08f
#define PIF  3.14159265358979323846f

// spherical-harmonic normalization constants
#define Y_C0 0.28209479177387814f   // 0.5*sqrt(1/pi)
#define Y_C1 0.4886025119029199f    // sqrt(3/(4pi))
#define Y_A2 1.0925484305920792f    // 0.5*sqrt(15/pi)
#define Y_B2 0.31539156525252005f   // 0.25*sqrt(5/pi)
#define Y_C2 0.5462742152960396f    // 0.25*sqrt(15/pi)
#define Y_A3 0.5900435899266435f    // 0.25*sqrt(35/(2pi))
#define Y_B3 2.8906114426405543f    // 0.5*sqrt(105/pi)
#define Y_C3 0.4570457994644657f    // 0.25*sqrt(21/(2pi))
#define Y_D3 0.3731763325901154f    // 0.25*sqrt(7/pi)
#define Y_E3 1.445305610320144f     // 0.25*sqrt(105/pi)

// ---------------------------------------------------------------------------
// Phase 1: sphc[b,i,c] = sum_j phi(d_ij) * Y_c(unit_ij) / C_safe
// One wave (32 lanes) per output row i; coords for the batch staged in LDS.
// ---------------------------------------------------------------------------
__global__ void __launch_bounds__(256)
sphc_kernel(const float* __restrict__ coords,
            const unsigned char* __restrict__ mask,
            float* __restrict__ sphc)
{
    __shared__ float cs[NN * 3];
    __shared__ unsigned char ms[NN];

    const int tid   = threadIdx.x;
    const int grow0 = blockIdx.x * 8;          // 8 rows per block (8 waves)
    const int b     = grow0 / NN;              // blocks never straddle batches

    for (int t = tid; t < NN; t += 256) {
        cs[3 * t + 0] = coords[((size_t)(b * NN + t)) * 3 + 0];
        cs[3 * t + 1] = coords[((size_t)(b * NN + t)) * 3 + 1];
        cs[3 * t + 2] = coords[((size_t)(b * NN + t)) * 3 + 2];
        ms[t] = mask[b * NN + t];
    }
    __syncthreads();

    const int wave = tid >> 5;
    const int lane = tid & 31;
    const int irow = (grow0 % NN) + wave;

    const float xi = cs[3 * irow + 0];
    const float yi = cs[3 * irow + 1];
    const float zi = cs[3 * irow + 2];
    const bool  mi = (ms[irow] != 0);

    float acc[17];
#pragma unroll
    for (int k = 0; k < 17; ++k) acc[k] = 0.0f;

    for (int j = lane; j < NN; j += 32) {
        const float dx = cs[3 * j + 0] - xi;
        const float dy = cs[3 * j + 1] - yi;
        const float dz = cs[3 * j + 2] - zi;
        const float sq = dx * dx + dy * dy + dz * dz;
        const float d  = (sq > 0.0f) ? sqrtf(sq) : 0.0f;
        const float iv = 1.0f / (d + EPSV);
        const float x = dx * iv, y = dy * iv, z = dz * iv;

        const bool nb = mi && (ms[j] != 0) && (j != irow);
        const float phi = (nb && d < RCUT)
                        ? 0.5f * (__cosf(PIF * d / RCUT) + 1.0f) : 0.0f;

        const float x2 = x * x, y2 = y * y, z2 = z * z;
        acc[0]  += phi;
        acc[1]  += phi * Y_C0;                               // l=0
        acc[2]  += phi * (Y_C1 * y);                         // l=1
        acc[3]  += phi * (Y_C1 * z);
        acc[4]  += phi * (Y_C1 * x);
        acc[5]  += phi * (Y_A2 * x * y);                     // l=2
        acc[6]  += phi * (Y_A2 * y * z);
        acc[7]  += phi * (Y_B2 * (3.0f * z2 - 1.0f));
        acc[8]  += phi * (Y_A2 * x * z);
        acc[9]  += phi * (Y_C2 * (x2 - y2));
        acc[10] += phi * (Y_A3 * y * (3.0f * x2 - y2));      // l=3
        acc[11] += phi * (Y_B3 * x * y * z);
        acc[12] += phi * (Y_C3 * y * (5.0f * z2 - 1.0f));
        acc[13] += phi * (Y_D3 * (5.0f * z2 * z - 3.0f * z));
        acc[14] += phi * (Y_C3 * x * (5.0f * z2 - 1.0f));
        acc[15] += phi * (Y_E3 * (x2 - y2) * z);
        acc[16] += phi * (Y_A3 * x * (x2 - 3.0f * y2));
    }

    // wave32 butterfly reduction
#pragma unroll
    for (int off = 16; off > 0; off >>= 1) {
#pragma unroll
        for (int k = 0; k < 17; ++k)
            acc[k] += __shfl_xor(acc[k], off, 32);
    }

    if (lane == 0) {
        const float C     = acc[0];
        const bool  small = (C < EPSV);
        const float cinv  = 1.0f / (small ? 1.0f : C);
        float* row = sphc + ((size_t)(b * NN + irow)) * NC;
#pragma unroll
        for (int c = 0; c < NC; ++c)
            row[c] = small ? 0.0f : acc[1 + c] * cinv;
    }
}

// ---------------------------------------------------------------------------
// Phase 2: X[i,j] = ||chi_i - chi_j||, row softmax, (1 - xr)^3 cutoff (P=1).
// Gram matrix chi * chi^T computed with V_WMMA_F32_16X16X4_F32 (K=16 -> 4x).
// One block per (batch, 16-row tile): 8 waves x 6 Gram tiles each.
// ---------------------------------------------------------------------------
__global__ void __launch_bounds__(256)
xcut_kernel(const float* __restrict__ chi,
            const unsigned char* __restrict__ mask,
            float* __restrict__ outX)
{
    __shared__ float norm_s[NN];        // ||chi_j||^2
    __shared__ float X_s[16 * NN];      // distance row-block (48 KB)
    __shared__ int   nv_s;

    const int tid  = threadIdx.x;
    const int lane = tid & 31;
    const int wave = tid >> 5;
    const int b    = blockIdx.x / (NN / 16);
    const int ti   = blockIdx.x % (NN / 16);
    const int i0   = ti * 16;

    const float* chib = chi + (size_t)b * NN * NC;

    if (tid == 0) nv_s = 0;
    __syncthreads();

    int pm = 0;
    for (int t = tid; t < NN; t += 256) {
        float s = 0.0f;
#pragma unroll
        for (int c = 0; c < NC; ++c) {
            const float v = chib[t * NC + c];
            s += v * v;
        }
        norm_s[t] = s;
        pm += (mask[b * NN + t] != 0) ? 1 : 0;
    }
    atomicAdd(&nv_s, pm);
    __syncthreads();

    // A fragment: 16x4 f32 per ISA layout — lanes 0-15: K=0,1; lanes 16-31: K=2,3
    const int mrow = lane & 15;
    const int koff = (lane >= 16) ? 2 : 0;
    v2f afrag[4];
#pragma unroll
    for (int k = 0; k < 4; ++k) {
        const float* p = chib + (size_t)(i0 + mrow) * NC + 4 * k + koff;
        afrag[k].x = p[0];
        afrag[k].y = p[1];
    }

    for (int tj = wave; tj < NN / 16; tj += 8) {
        v2f bfrag[4];
#pragma unroll
        for (int k = 0; k < 4; ++k) {
            const float* p = chib + (size_t)(tj * 16 + mrow) * NC + 4 * k + koff;
            bfrag[k].x = p[0];
            bfrag[k].y = p[1];
        }

        v8f g = {0.0f, 0.0f, 0.0f, 0.0f, 0.0f, 0.0f, 0.0f, 0.0f};
#pragma unroll
        for (int k = 0; k < 4; ++k) {
            // D = A x B + C ; fp32 16x16 tile of the Gram matrix, K advanced by 4
            g = __builtin_amdgcn_wmma_f32_16x16x4_f32(
                    /*neg_a=*/false, afrag[k],
                    /*neg_b=*/false, bfrag[k],
                    /*c_mod=*/(short)0, g,
                    /*reuse_a=*/false, /*reuse_b=*/false);
        }

        // C/D layout: VGPR r -> M = r (lanes 0-15) or r+8 (lanes 16-31), N = lane&15
#pragma unroll
        for (int r = 0; r < 8; ++r) {
            const int m = (lane < 16) ? r : (r + 8);
            const int j = tj * 16 + mrow;
            float s = norm_s[i0 + m] + norm_s[j] - 2.0f * g[r];
            X_s[m * NN + j] = (s > 0.0f) ? sqrtf(s) : 0.0f;
        }
    }
    __syncthreads();

    const float nvf = (float)nv_s;   // xr = softmax / (KAPPA / n_valid), KAPPA = 1

    // 2 rows per wave: stable softmax + polynomial cutoff (P=1 -> (1-xr)^3 form)
    for (int m = wave * 2; m < wave * 2 + 2; ++m) {
        const float* xr_row = &X_s[m * NN];

        float mx = -3.0e38f;
        for (int j = lane; j < NN; j += 32) mx = fmaxf(mx, xr_row[j]);
#pragma unroll
        for (int off = 16; off > 0; off >>= 1)
            mx = fmaxf(mx, __shfl_xor(mx, off, 32));

        float sum = 0.0f;
        for (int j = lane; j < NN; j += 32) sum += __expf(xr_row[j] - mx);
#pragma unroll
        for (int off = 16; off > 0; off >>= 1)
            sum += __shfl_xor(sum, off, 32);
        const float isum = 1.0f / sum;

        float* orow = outX + ((size_t)(b * NN + i0 + m)) * NN;
        for (int j = lane; j < NN; j += 32) {
            const float soft = __expf(xr_row[j] - mx) * isum;
            const float xr   = soft * nvf;
            orow[j] = 1.0f - 3.0f * xr + 3.0f * xr * xr - xr * xr * xr;
        }
    }
}

extern "C" void kernel_launch(void* const* d_in, const int* in_sizes, int n_in,
                              void* d_out, int out_size, void* d_ws, size_t ws_size,
                              hipStream_t stream) {
    const float*         coords = (const float*)d_in[0];         // [8,768,3] f32
    const unsigned char* mask   = (const unsigned char*)d_in[1]; // [8,768] bool
    float* out  = (float*)d_out;
    float* sphc = out;                         // [8,768,16]
    float* outX = out + (size_t)NB * NN * NC;  // [8,768,768]

    sphc_kernel<<<NB * NN / 8, 256, 0, stream>>>(coords, mask, sphc);
    xcut_kernel<<<NB * (NN / 16), 256, 0, stream>>>(sphc, mask, outX);
}